// RNN_52055003628080
// MI455X (gfx1250) — compile-verified
//
#include <hip/hip_runtime.h>
#include <math.h>

typedef __attribute__((ext_vector_type(2))) float v2f;
typedef __attribute__((ext_vector_type(8))) float v8f;

// Problem dims (fixed by the reference).
constexpr int T_DIM = 512;
constexpr int B_DIM = 64;
constexpr int I_DIM = 512;
constexpr int H_DIM = 512;

// ---- Phase 1 (xproj GEMM) tiling ----
constexpr int BM  = 64;   // M rows per block
constexpr int BN  = 64;   // N cols per block
constexpr int BK  = 32;   // K per LDS stage
constexpr int LS1 = 36;   // padded LDS row stride (floats): 144B, 16B-aligned, conflict-free
constexpr int KSTAGES = I_DIM / BK;

// ---- Phase 2 (recurrent scan) tiling ----
constexpr int NBLK2 = 16; // persistent workgroups; block b owns H columns [32b, 32b+32)
constexpr int LS2   = 516; // padded LDS row stride: 2064B, 16B-aligned, conflict-free

// =====================================================================
// CDNA5 async global->LDS helpers (ASYNCcnt-tracked; see CDNA5 ISA).
// ROCm clang declares the builtin with (addrspace(1) int4*, addrspace(3) int4*,
// imm offset, imm cpol) — per the round-2 diagnostic, pointees are gcc-style
// 16-byte int vectors and AS1 prints as `__device__`.
// =====================================================================
typedef int v4i_vs __attribute__((vector_size(16)));
typedef __attribute__((address_space(1))) v4i_vs as1_v4i;
typedef __attribute__((address_space(3))) v4i_vs as3_v4i;

__device__ __forceinline__ void async_copy_b128(const float* g, float* l) {
#if defined(__HIP_DEVICE_COMPILE__)
#if __has_builtin(__builtin_amdgcn_global_load_async_to_lds_b128)
  __builtin_amdgcn_global_load_async_to_lds_b128(
      (as1_v4i*)(unsigned long long)(const void*)g,
      (as3_v4i*)(unsigned)(unsigned long long)(void*)l,
      0, 0);
#else
  // Flat->local addrspacecast == truncate to low 32 bits (LDS byte offset).
  const unsigned lds_addr = (unsigned)(unsigned long long)(void*)l;
  asm volatile("global_load_async_to_lds_b128 %0, %1, off"
               :: "v"(lds_addr), "v"(g)
               : "memory");
#endif
#else
  (void)g; (void)l;
#endif
}

__device__ __forceinline__ void wait_async0() {
#if defined(__HIP_DEVICE_COMPILE__)
#if __has_builtin(__builtin_amdgcn_s_wait_asynccnt)
  __builtin_amdgcn_s_wait_asynccnt(0);
#else
  asm volatile("s_wait_asynccnt 0x0" ::: "memory");
#endif
#endif
}

// =====================================================================
// Phase 1: xproj[t,b,:] = x[t,b,:] @ w_ih^T + b_ih + b_hh, written to d_out.
// Exact fp32 via V_WMMA_F32_16X16X4_F32. 8 waves/block: 4 (M) x 2 (N).
// K-stages double-buffered: async-copy stage s+1 while WMMAs run on stage s.
// =====================================================================
__global__ __launch_bounds__(256) void xproj_wmma(
    const float* __restrict__ x, const float* __restrict__ w_ih,
    const float* __restrict__ b_ih, const float* __restrict__ b_hh,
    float* __restrict__ out) {
  __shared__ float Xs[2][BM * LS1];
  __shared__ float Ws[2][BN * LS1];

  const int tid  = threadIdx.x;
  const int lane = tid & 31;
  const int wid  = tid >> 5;
  const int half = lane >> 4;  // 0: K pair {0,1}; 1: K pair {2,3}
  const int lr   = lane & 15;
  const int wm   = wid & 3;    // M subtile 0..3
  const int wn   = wid >> 2;   // N subtile pair 0..1
  const long m0  = (long)blockIdx.y * BM;
  const int  n0  = blockIdx.x * BN;

  const int row0 = tid >> 3;   // 0..31 (i-loop adds 32)
  const int c4   = tid & 7;    // float4 column within the 32-float row

  auto issue_stage = [&](int k0, int buf) {
#pragma unroll
    for (int i = 0; i < 2; ++i) {
      const int r = row0 + i * 32;
      async_copy_b128(&x[(m0 + r) * (long)I_DIM + k0 + c4 * 4],
                      &Xs[buf][r * LS1 + c4 * 4]);
      async_copy_b128(&w_ih[(long)(n0 + r) * I_DIM + k0 + c4 * 4],
                      &Ws[buf][r * LS1 + c4 * 4]);
    }
  };

  v8f c0 = {};
  v8f c1 = {};

  issue_stage(0, 0);
  for (int s = 0; s < KSTAGES; ++s) {
    wait_async0();     // my async copies for buffer s&1 are in LDS
    __syncthreads();   // everyone's copies done; stage s-1 reads all finished
    if (s + 1 < KSTAGES) issue_stage((s + 1) * BK, (s + 1) & 1);

    const float* Xb = Xs[s & 1];
    const float* Wb = Ws[s & 1];
#pragma unroll
    for (int kk = 0; kk < BK; kk += 4) {
      const int ks = kk + half * 2;
      v2f a  = *(const v2f*)&Xb[(wm * 16 + lr) * LS1 + ks];
      v2f b0 = *(const v2f*)&Wb[(wn * 32 + lr) * LS1 + ks];
      v2f b1 = *(const v2f*)&Wb[(wn * 32 + 16 + lr) * LS1 + ks];
      c0 = __builtin_amdgcn_wmma_f32_16x16x4_f32(false, a, false, b0,
                                                 (short)0, c0, false, false);
      c1 = __builtin_amdgcn_wmma_f32_16x16x4_f32(false, a, false, b1,
                                                 (short)0, c1, false, false);
    }
  }

  // Epilogue: C/D layout — VGPR j: lanes 0-15 -> M=j, lanes 16-31 -> M=j+8; N=lr.
  const int nc0 = n0 + wn * 32 + lr;
  const int nc1 = nc0 + 16;
  const float bias0 = b_ih[nc0] + b_hh[nc0];
  const float bias1 = b_ih[nc1] + b_hh[nc1];
#pragma unroll
  for (int j = 0; j < 8; ++j) {
    const long m = m0 + wm * 16 + half * 8 + j;
    out[m * H_DIM + nc0] = c0[j] + bias0;
    out[m * H_DIM + nc1] = c1[j] + bias1;
  }
}

// =====================================================================
// Phase 2: persistent-grid recurrent scan with grid-wide barrier per step.
// =====================================================================
__device__ __forceinline__ void grid_sync(unsigned* cnt, unsigned* gen,
                                          unsigned nb) {
  __threadfence();   // release our h_new / output[t] writes to device scope
  __syncthreads();
  if (threadIdx.x == 0) {
    const unsigned old =
        __hip_atomic_load(gen, __ATOMIC_ACQUIRE, __HIP_MEMORY_SCOPE_AGENT);
    const unsigned prev =
        __hip_atomic_fetch_add(cnt, 1u, __ATOMIC_ACQ_REL, __HIP_MEMORY_SCOPE_AGENT);
    if (prev == nb - 1u) {
      __hip_atomic_store(cnt, 0u, __ATOMIC_RELAXED, __HIP_MEMORY_SCOPE_AGENT);
      __hip_atomic_fetch_add(gen, 1u, __ATOMIC_ACQ_REL, __HIP_MEMORY_SCOPE_AGENT);
    } else {
      while (__hip_atomic_load(gen, __ATOMIC_ACQUIRE,
                               __HIP_MEMORY_SCOPE_AGENT) == old) {
        __builtin_amdgcn_s_sleep(2);
      }
    }
  }
  __syncthreads();
  // Subsequent h reads must not be served from stale WGP cache lines.
  __builtin_amdgcn_fence(__ATOMIC_ACQUIRE, "agent");
}

__global__ __launch_bounds__(256) void rnn_scan_wmma(
    const float* __restrict__ w_hh, float* __restrict__ out,
    float* __restrict__ hbuf, unsigned* __restrict__ bar,
    float* __restrict__ hn_out) {
  extern __shared__ float lds[];
  float* Wsh = lds;             // 32 x LS2 : this block's w_hh rows, resident
  float* Hs  = lds + 32 * LS2;  // 64 x LS2 : h_{t-1} staged per step

  const int tid  = threadIdx.x;
  const int lane = tid & 31;
  const int wid  = tid >> 5;
  const int half = lane >> 4;
  const int lr   = lane & 15;
  const int wm   = wid & 3;          // M subtile (B rows 16*wm..)
  const int wn   = wid >> 2;         // N subtile 0..1
  const int n0   = blockIdx.x * 32;  // 16 blocks x 32 columns = H
  const int m0w  = wm * 16;
  const int ncol = n0 + wn * 16 + lr;

  // Preload w_hh[n0:n0+32, :] into LDS once (resident for all 512 steps).
  for (int f = tid; f < 32 * 128; f += 256) {
    const int r  = f >> 7;
    const int cw = f & 127;
    async_copy_b128(&w_hh[(long)(n0 + r) * H_DIM + cw * 4],
                    &Wsh[r * LS2 + cw * 4]);
  }

  float* hA = hbuf;
  float* hB = hbuf + B_DIM * H_DIM;

  for (int t = 0; t < T_DIM; ++t) {
    const float* hc = (t & 1) ? hB : hA;
    float*       hn = (t & 1) ? hA : hB;

    // Stage full h_{t-1} [64 x 512] into LDS via async copies (32/thread;
    // ASYNCcnt max 48 outstanding on step 0 -- under the 6-bit limit).
    for (int f = tid; f < 64 * 128; f += 256) {
      const int r  = f >> 7;
      const int ch = f & 127;
      async_copy_b128(&hc[r * H_DIM + ch * 4], &Hs[r * LS2 + ch * 4]);
    }
    wait_async0();
    __syncthreads();

    v8f c = {};
#pragma unroll 8
    for (int kk = 0; kk < H_DIM; kk += 4) {
      const int ks = kk + half * 2;
      v2f a = *(const v2f*)&Hs[(m0w + lr) * LS2 + ks];
      v2f b = *(const v2f*)&Wsh[(wn * 16 + lr) * LS2 + ks];
      c = __builtin_amdgcn_wmma_f32_16x16x4_f32(false, a, false, b,
                                                (short)0, c, false, false);
    }

    // h_t = tanh(xproj_t + h_{t-1} @ W_hh^T); overwrite output[t] in place.
    float* ot = out + (long)t * (B_DIM * H_DIM);
#pragma unroll
    for (int j = 0; j < 8; ++j) {
      const int m = m0w + half * 8 + j;
      const float v = tanhf(ot[m * H_DIM + ncol] + c[j]);
      ot[m * H_DIM + ncol] = v;
      hn[m * H_DIM + ncol] = v;
      if (t == T_DIM - 1) hn_out[m * H_DIM + ncol] = v;
    }

    grid_sync(&bar[0], &bar[1], (unsigned)gridDim.x);
  }
}

// Zero h0 and the barrier words (d_ws is poisoned by the harness).
__global__ void init_state(float* h0, unsigned* bar) {
  const int i = blockIdx.x * blockDim.x + threadIdx.x;
  if (i < B_DIM * H_DIM) h0[i] = 0.0f;
  if (i == 0) { bar[0] = 0u; bar[1] = 0u; }
}

extern "C" void kernel_launch(void* const* d_in, const int* in_sizes, int n_in,
                              void* d_out, int out_size, void* d_ws,
                              size_t ws_size, hipStream_t stream) {
  const float* x    = (const float*)d_in[0];
  const float* w_ih = (const float*)d_in[1];
  const float* w_hh = (const float*)d_in[2];
  const float* b_ih = (const float*)d_in[3];
  const float* b_hh = (const float*)d_in[4];
  float* out = (float*)d_out;  // [T*B*H] output, then [B*H] h_n

  float*    hbuf = (float*)d_ws;  // 2 * B * H floats (h ping-pong)
  unsigned* bar =
      (unsigned*)((char*)d_ws + (size_t)2 * B_DIM * H_DIM * sizeof(float));

  init_state<<<(B_DIM * H_DIM + 255) / 256, 256, 0, stream>>>(hbuf, bar);

  dim3 g1(H_DIM / BN, (T_DIM * B_DIM) / BM);  // (8, 512)
  xproj_wmma<<<g1, 256, 0, stream>>>(x, w_ih, b_ih, b_hh, out);

  const size_t lds2 = (size_t)(32 + 64) * LS2 * sizeof(float);  // ~194 KB
  (void)hipFuncSetAttribute(reinterpret_cast<const void*>(rnn_scan_wmma),
                            hipFuncAttributeMaxDynamicSharedMemorySize,
                            (int)lds2);
  float* hn_out = out + (size_t)T_DIM * B_DIM * H_DIM;
  rnn_scan_wmma<<<NBLK2, 256, lds2, stream>>>(w_hh, out, hbuf, bar, hn_out);
}